// ClassAwareLoss_63359357551371
// MI455X (gfx1250) — compile-verified
//
#include <hip/hip_runtime.h>

#define NC   100
#define NB   16384
#define DIM  256
#define FT   1600

typedef __attribute__((ext_vector_type(2))) float v2f;
typedef __attribute__((ext_vector_type(4))) float v4f;
typedef __attribute__((ext_vector_type(8))) float v8f;

// ---------------- kernel 0: zero accumulators & histograms ----------------
__global__ void k0_zero(float* acc, int* class_count, int* fcount) {
    int i = threadIdx.x;
    if (i < 2)  acc[i] = 0.0f;          // acc[0]=caloss, acc[1]=reg
    if (i < NC) { class_count[i] = 0; fcount[i] = 0; }
}

// ---------------- kernel 1: class histogram + frame ranges ----------------
__global__ void k1_hist(const int* __restrict__ target,
                        const int* __restrict__ frame_class,
                        int* class_count, int* fstart, int* fcount) {
    int i = blockIdx.x * blockDim.x + threadIdx.x;
    if (i < NB) atomicAdd(&class_count[target[i]], 1);
    if (i < FT) {
        int c = frame_class[i];
        atomicAdd(&fcount[c], 1);
        if (i == 0 || frame_class[i - 1] != c) fstart[c] = i;  // contiguous runs
    }
}

// ---------------- kernel 2: tiny serial exclusive scan (nc=100) -----------
__global__ void k2_scan(const int* __restrict__ class_count,
                        int* class_start, int* cursor) {
    if (threadIdx.x == 0 && blockIdx.x == 0) {
        int s = 0;
        for (int c = 0; c < NC; ++c) {
            class_start[c] = s;
            cursor[c]      = s;
            s += class_count[c];
        }
        class_start[NC] = s;
    }
}

// ------- kernel 3: per-row norm (1 wave/row), reg accum, class scatter ----
__global__ __launch_bounds__(256) void k3_norm(const float* __restrict__ x,
                                               const int* __restrict__ target,
                                               float* inv_norm, int* rowlist,
                                               int* cursor, float* acc) {
    int lane = threadIdx.x & 31;
    int wave = threadIdx.x >> 5;
    int row  = blockIdx.x * 8 + wave;          // grid = NB/8 blocks
    const float* p = x + (size_t)row * DIM + lane * 8;
    v4f a = *(const v4f*)p;
    v4f b = *(const v4f*)(p + 4);
    float s = a.x * a.x + a.y * a.y + a.z * a.z + a.w * a.w +
              b.x * b.x + b.y * b.y + b.z * b.z + b.w * b.w;
    #pragma unroll
    for (int off = 16; off > 0; off >>= 1) s += __shfl_xor(s, off, 32);
    if (lane == 0) {
        float norm = sqrtf(s);
        norm = fmaxf(norm, 1e-08f);
        inv_norm[row] = 1.0f / norm;
        float d = norm - 1.0f;
        atomicAdd(&acc[1], d * d);             // reg
        int t   = target[row];
        int pos = atomicAdd(&cursor[t], 1);
        rowlist[pos] = row;
    }
}

// ------- kernel 4: per-class 16x16 WMMA tiles, fp32 16x16x4 ---------------
__global__ __launch_bounds__(32) void k4_wmma(const float* __restrict__ x,
                                              const float* __restrict__ frames,
                                              const float* __restrict__ cosine_c,
                                              const int* __restrict__ rowlist,
                                              const float* __restrict__ inv_norm,
                                              const int* __restrict__ class_start,
                                              const int* __restrict__ class_count,
                                              const int* __restrict__ fstart,
                                              const int* __restrict__ fcount,
                                              float* acc) {
    const int c      = blockIdx.y;
    const int mtile  = blockIdx.x;
    const int ntile  = blockIdx.z;
    const int ccount = class_count[c];
    const int fc     = fcount[c];
    if (mtile * 16 >= ccount || ntile * 16 >= fc) return;   // wave-uniform exit

    const int cstart = class_start[c];
    const int fs     = fstart[c];
    const int lane   = threadIdx.x;
    const int half   = lane >> 4;       // K-half selector for A/B layouts
    const int ml     = lane & 15;       // M for A-lane, N for B-lane

    // A operand: 16 gathered, normalized batch rows of this class
    const int  mIdx   = mtile * 16 + ml;
    const bool mValid = mIdx < ccount;
    const int  row    = rowlist[cstart + (mValid ? mIdx : 0)];
    const float ainv  = mValid ? inv_norm[row] : 0.0f;      // zero out pad rows
    const float* aptr = x + (size_t)row * DIM + half * 2;

    // B operand: 16 frames of this class
    const int  nIdx   = ntile * 16 + ml;
    const bool nValid = nIdx < fc;
    const int  fr     = fs + (nValid ? nIdx : 0);
    const float* bptr = frames + (size_t)fr * DIM + half * 2;

    v8f accm = {};
    #pragma unroll 4
    for (int k = 0; k < DIM; k += 4) {
        v2f av, bv;
        av.x = aptr[k] * ainv;          // A[16x4]: lane=M, VGPR+half = K
        av.y = aptr[k + 1] * ainv;
        bv.x = bptr[k];                 // B[4x16]: lane=N, VGPR+half = K
        bv.y = bptr[k + 1];
        accm = __builtin_amdgcn_wmma_f32_16x16x4_f32(
                   false, av, false, bv, (short)0, accm, false, false);
    }

    // C layout: VGPR i -> M = i + half*8, N = ml. Masked (1-dot)^2 sum.
    const float cw = cosine_c[c];
    float s = 0.0f;
    #pragma unroll
    for (int i = 0; i < 8; ++i) {
        int  M  = i + half * 8;
        bool ok = nValid && (mtile * 16 + M < ccount);
        if (ok) {
            float d = 1.0f - accm[i];
            s += d * d;
        }
    }
    s *= cw;
    #pragma unroll
    for (int off = 16; off > 0; off >>= 1) s += __shfl_xor(s, off, 32);
    if (lane == 0) atomicAdd(&acc[0], s);
}

// ---------------- kernel 5: finalize scalar -------------------------------
__global__ void k5_final(const float* __restrict__ acc, float* out) {
    if (threadIdx.x == 0 && blockIdx.x == 0)
        out[0] = (acc[0] + 0.0006f * acc[1]) * (1.0f / (float)NB);
}

extern "C" void kernel_launch(void* const* d_in, const int* in_sizes, int n_in,
                              void* d_out, int out_size, void* d_ws, size_t ws_size,
                              hipStream_t stream) {
    const float* x           = (const float*)d_in[0];   // [NB, DIM]
    const float* frames      = (const float*)d_in[1];   // [FT, DIM]
    const float* cosine_c    = (const float*)d_in[2];   // [NC]
    const int*   target      = (const int*)d_in[3];     // [NB]
    const int*   frame_class = (const int*)d_in[4];     // [FT]
    float*       out         = (float*)d_out;

    // workspace layout (all 4-byte words)
    float* acc         = (float*)d_ws;            // [2]  caloss, reg
    int*   class_count = (int*)(acc + 2);         // [NC]
    int*   class_start = class_count + NC;        // [NC+1]
    int*   cursor      = class_start + NC + 1;    // [NC]
    int*   fstart      = cursor + NC;             // [NC]
    int*   fcount      = fstart + NC;             // [NC]
    float* inv_norm    = (float*)(fcount + NC);   // [NB]
    int*   rowlist     = (int*)(inv_norm + NB);   // [NB]

    k0_zero<<<1, 256, 0, stream>>>(acc, class_count, fcount);
    k1_hist<<<(NB + 255) / 256, 256, 0, stream>>>(target, frame_class,
                                                  class_count, fstart, fcount);
    k2_scan<<<1, 32, 0, stream>>>(class_count, class_start, cursor);
    k3_norm<<<NB / 8, 256, 0, stream>>>(x, target, inv_norm, rowlist, cursor, acc);

    dim3 grid4(NB / 16, NC, 2);   // m-tiles (worst case), classes, n-tiles (fc<=31)
    k4_wmma<<<grid4, 32, 0, stream>>>(x, frames, cosine_c, rowlist, inv_norm,
                                      class_start, class_count, fstart, fcount, acc);
    k5_final<<<1, 32, 0, stream>>>(acc, out);
}